// VRFC_5059471474718
// MI455X (gfx1250) — compile-verified
//
#include <hip/hip_runtime.h>
#include <hip/hip_bf16.h>

#define N_OBJ   1024
#define C_CLS   151
#define RELD    4096
#define M_REL   4096
#define NUM_REL 51
#define NMS_TH  0.3f

#define KCHUNK  64     // K-floats staged per chunk
#define APITCH  68     // LDS row pitch (floats): 4 mod 64 -> conflict-free frag reads

typedef __attribute__((ext_vector_type(2))) float v2f;
typedef __attribute__((ext_vector_type(8))) float v8f;

// ---------------------------------------------------------------------------
// 1) obj_dists2 = obj_logits (coalesced copy)
// ---------------------------------------------------------------------------
__global__ void copy_logits_kernel(const float* __restrict__ in,
                                   float* __restrict__ out, int n) {
    int i = blockIdx.x * blockDim.x + threadIdx.x;
    if (i < n) out[i] = in[i];
}

// ---------------------------------------------------------------------------
// 2a) row softmax (one row per thread)
// ---------------------------------------------------------------------------
__global__ void softmax_rows_kernel(const float* __restrict__ logits,
                                    float* __restrict__ probs) {
    int row = blockIdx.x * blockDim.x + threadIdx.x;
    if (row >= N_OBJ) return;
    const float* r = logits + (size_t)row * C_CLS;
    float m = r[0];
    for (int c = 1; c < C_CLS; ++c) m = fmaxf(m, r[c]);
    float s = 0.f;
    for (int c = 0; c < C_CLS; ++c) s += __expf(r[c] - m);
    float inv = 1.f / s;
    float* p = probs + (size_t)row * C_CLS;
    for (int c = 0; c < C_CLS; ++c) p[c] = __expf(r[c] - m) * inv;
}

// ---------------------------------------------------------------------------
// 2b) per-class greedy NMS (one 1024-thread workgroup per class)
// ---------------------------------------------------------------------------
__global__ __launch_bounds__(1024) void nms_kernel(
        const float* __restrict__ probs,
        const float* __restrict__ boxes,   // [N][C][4]
        float* __restrict__ masked) {      // [N][C-1]
    __shared__ float s_sc[N_OBJ];
    __shared__ int   s_id[N_OBJ];
    __shared__ float sx1[N_OBJ], sy1[N_OBJ], sx2[N_OBJ], sy2[N_OBJ], sar[N_OBJ];
    __shared__ int   keep[N_OBJ];

    const int tid = threadIdx.x;
    const int cls = blockIdx.x + 1;        // classes 1..150

    s_sc[tid] = probs[(size_t)tid * C_CLS + cls];
    s_id[tid] = tid;
    __syncthreads();

    // Bitonic sort, descending by score (carry original index).
    for (int k = 2; k <= N_OBJ; k <<= 1) {
        for (int j = k >> 1; j > 0; j >>= 1) {
            int ixj = tid ^ j;
            if (ixj > tid) {
                bool descBlock = ((tid & k) == 0);
                float a = s_sc[tid], b = s_sc[ixj];
                bool doSwap = descBlock ? (a < b) : (a > b);
                if (doSwap) {
                    s_sc[tid] = b; s_sc[ixj] = a;
                    int t = s_id[tid]; s_id[tid] = s_id[ixj]; s_id[ixj] = t;
                }
            }
            __syncthreads();
        }
    }

    const int n = s_id[tid];
    const float* bx = boxes + ((size_t)n * C_CLS + cls) * 4;
    float x1 = bx[0], y1 = bx[1], x2 = bx[2], y2 = bx[3];
    sx1[tid] = x1; sy1[tid] = y1; sx2[tid] = x2; sy2[tid] = y2;
    sar[tid] = (x2 - x1) * (y2 - y1);
    keep[tid] = 1;
    __syncthreads();

    for (int i = 0; i < N_OBJ; ++i) {
        if (keep[i]) {
            if (tid > i && keep[tid]) {
                float ix1 = fmaxf(sx1[i], sx1[tid]);
                float iy1 = fmaxf(sy1[i], sy1[tid]);
                float ix2 = fminf(sx2[i], sx2[tid]);
                float iy2 = fminf(sy2[i], sy2[tid]);
                float iw = fmaxf(ix2 - ix1, 0.f);
                float ih = fmaxf(iy2 - iy1, 0.f);
                float inter = iw * ih;
                float iou = inter / (sar[i] + sar[tid] - inter + 1e-10f);
                if (iou > NMS_TH) keep[tid] = 0;
            }
        }
        __syncthreads();
    }

    masked[(size_t)n * (C_CLS - 1) + (cls - 1)] =
        keep[tid] ? probs[(size_t)n * C_CLS + cls] : 0.f;
}

// ---------------------------------------------------------------------------
// 2c) obj_preds[row] = argmax_c(masked[row]) + 1
// ---------------------------------------------------------------------------
__global__ void argmax_kernel(const float* __restrict__ masked,
                              float* __restrict__ preds) {
    int row = blockIdx.x * blockDim.x + threadIdx.x;
    if (row >= N_OBJ) return;
    const float* r = masked + (size_t)row * (C_CLS - 1);
    float best = r[0];
    int bi = 0;
    for (int c = 1; c < C_CLS - 1; ++c) {
        float v = r[c];
        if (v > best) { best = v; bi = c; }
    }
    preds[row] = (float)(bi + 1);
}

// ---------------------------------------------------------------------------
// 3) rel_dists = vr @ W^T + b   via V_WMMA_F32_16X16X4_F32
//    Block = 128 threads (4 waves) owns a 16-row M tile. A tile staged into
//    LDS (double-buffered) with global_load_async_to_lds_b128; wave w computes
//    output columns [16w, 16w+16). Invalid B columns (>=51) read clamped rows;
//    garbage only reaches D columns that are never stored.
// ---------------------------------------------------------------------------
__global__ __launch_bounds__(128) void rel_gemm_wmma_kernel(
        const float* __restrict__ vr,     // [M][K]
        const float* __restrict__ W,      // [NUM_REL][K]
        const float* __restrict__ bias,   // [NUM_REL]
        float* __restrict__ out) {        // [M][NUM_REL]
    __shared__ __align__(16) float sA[2][16 * APITCH];

    const int tid  = threadIdx.x;
    const int lane = tid & 31;
    const int wave = tid >> 5;                // 0..3 -> column tile
    const int m0   = blockIdx.x * 16;
    const int r0   = wave * 16;
    const int lm   = lane & 15;
    const int hi   = lane >> 4;               // 0 or 1
    const int col  = r0 + lm;
    const bool colv = (col < NUM_REL);

    const float bval = colv ? bias[col] : 0.f;
    v8f acc = { bval, bval, bval, bval, bval, bval, bval, bval };

    // B row pointer, clamped to a legal row for invalid columns.
    const float* brow = W + (size_t)(colv ? col : (NUM_REL - 1)) * RELD + 2 * hi;

    // Cooperative A staging: 16 rows x KCHUNK floats = 256 float4 quanta;
    // thread t moves quanta t and t+128 (coalesced on the global side).
    const int q0 = tid, q1 = tid + 128;
    const int rA0 = q0 >> 4, cA0 = (q0 & 15) * 4;
    const int rA1 = q1 >> 4, cA1 = (q1 & 15) * 4;
    const float* g0 = vr + (size_t)(m0 + rA0) * RELD + cA0;
    const float* g1 = vr + (size_t)(m0 + rA1) * RELD + cA1;
    unsigned l0[2], l1[2];
    l0[0] = (unsigned)(uintptr_t)&sA[0][rA0 * APITCH + cA0];
    l0[1] = (unsigned)(uintptr_t)&sA[1][rA0 * APITCH + cA0];
    l1[0] = (unsigned)(uintptr_t)&sA[0][rA1 * APITCH + cA1];
    l1[1] = (unsigned)(uintptr_t)&sA[1][rA1 * APITCH + cA1];

#if defined(__HIP_DEVICE_COMPILE__)
#define STAGE_CHUNK(buf, kc)                                                   \
    do {                                                                       \
        unsigned long long ga0 = (unsigned long long)(uintptr_t)(g0 + (kc));   \
        unsigned long long ga1 = (unsigned long long)(uintptr_t)(g1 + (kc));   \
        asm volatile("global_load_async_to_lds_b128 %0, %1, off"               \
                     :: "v"(l0[buf]), "v"(ga0) : "memory");                    \
        asm volatile("global_load_async_to_lds_b128 %0, %1, off"               \
                     :: "v"(l1[buf]), "v"(ga1) : "memory");                    \
    } while (0)
#define WAIT_ASYNC() asm volatile("s_wait_asynccnt 0x0" ::: "memory")
#else
#define STAGE_CHUNK(buf, kc) do { (void)l0; (void)l1; (void)g0; (void)g1; } while (0)
#define WAIT_ASYNC() do { } while (0)
#endif

    STAGE_CHUNK(0, 0);
    WAIT_ASYNC();
    __syncthreads();

    const int nchunk = RELD / KCHUNK;
    for (int ch = 0; ch < nchunk; ++ch) {
        const int kc = ch * KCHUNK;
        if (ch + 1 < nchunk) STAGE_CHUNK((ch + 1) & 1, kc + KCHUNK);

        const float* sa = &sA[ch & 1][lm * APITCH + 2 * hi];
        const float* bp = brow + kc;
#pragma unroll
        for (int kk = 0; kk < KCHUNK; kk += 4) {
            v2f a = *(const v2f*)(sa + kk);     // ds_load_b64, conflict-free
            v2f b = *(const v2f*)(bp + kk);     // global_load_b64 along W row
#if defined(__HIP_DEVICE_COMPILE__)
            acc = __builtin_amdgcn_wmma_f32_16x16x4_f32(
                /*neg_a=*/false, a, /*neg_b=*/false, b,
                /*c_mod=*/(short)0, acc, /*reuse_a=*/false, /*reuse_b=*/false);
#else
            (void)a; (void)b;
#endif
        }
        WAIT_ASYNC();       // staging of next chunk complete (this wave)
        __syncthreads();    // all waves done reading current buf + staging next
    }

    // D layout: VGPR v, lane l -> row m0 + v + 8*(l>>4), col r0 + (l&15).
    if (colv) {
#pragma unroll
        for (int v = 0; v < 8; ++v) {
            int m = m0 + v + 8 * hi;
            out[(size_t)m * NUM_REL + col] = acc[v];
        }
    }
#undef STAGE_CHUNK
#undef WAIT_ASYNC
}

// ---------------------------------------------------------------------------
// launch
// ---------------------------------------------------------------------------
extern "C" void kernel_launch(void* const* d_in, const int* in_sizes, int n_in,
                              void* d_out, int out_size, void* d_ws, size_t ws_size,
                              hipStream_t stream) {
    (void)in_sizes; (void)n_in; (void)out_size; (void)ws_size;

    const float* obj_logits = (const float*)d_in[0];   // [N][C]
    const float* vr         = (const float*)d_in[1];   // [M][RELD]
    const float* boxes      = (const float*)d_in[2];   // [N][C][4]
    const float* W          = (const float*)d_in[3];   // [NUM_REL][RELD]
    const float* bias       = (const float*)d_in[4];   // [NUM_REL]

    float* out        = (float*)d_out;
    float* out_dists  = out;                               // N*C
    float* out_preds  = out + (size_t)N_OBJ * C_CLS;       // N
    float* out_rel    = out_preds + N_OBJ;                 // M*NUM_REL

    float* ws_probs   = (float*)d_ws;                      // N*C floats
    float* ws_masked  = ws_probs + (size_t)N_OBJ * C_CLS;  // N*(C-1) floats

    {
        int n = N_OBJ * C_CLS;
        copy_logits_kernel<<<(n + 255) / 256, 256, 0, stream>>>(obj_logits, out_dists, n);
    }
    softmax_rows_kernel<<<(N_OBJ + 255) / 256, 256, 0, stream>>>(obj_logits, ws_probs);
    nms_kernel<<<C_CLS - 1, N_OBJ, 0, stream>>>(ws_probs, boxes, ws_masked);
    argmax_kernel<<<(N_OBJ + 255) / 256, 256, 0, stream>>>(ws_masked, out_preds);
    rel_gemm_wmma_kernel<<<M_REL / 16, 128, 0, stream>>>(vr, W, bias, out_rel);
}